// _GraphConvSparse_89721866813830
// MI455X (gfx1250) — compile-verified
//
#include <hip/hip_runtime.h>

typedef __attribute__((ext_vector_type(2))) float v2f;
typedef __attribute__((ext_vector_type(8))) float v8f;

#define D 256  // D_IN == D_OUT == 256

// -------------------------------------------------------------------------
// Kernel 1: x = inputs @ W  via V_WMMA_F32_16X16X4_F32 (wave32)
// Block = 256 threads = 8 waves. Block covers 16 rows x 256 cols.
// Wave w owns cols [w*32, w*32+32): two 16x16 tiles, A-fragment reused.
// -------------------------------------------------------------------------
__global__ __launch_bounds__(256) void gemm_f32_wmma(
    const float* __restrict__ A,   // [nNodes, 256]
    const float* __restrict__ W,   // [256, 256]
    float* __restrict__ X,         // [nNodes, 256]
    int nNodes) {
  const int lane = threadIdx.x & 31;
  const int wave = threadIdx.x >> 5;
  const int l    = lane & 15;       // M (for A) / N (for B/C)
  const int half = lane >> 4;       // selects K sub-pair {0,1} vs {2,3}
  const int rowBase = blockIdx.x * 16;

  // clamp load row so WMMA runs with full EXEC even on a (hypothetical) tail
  int arowIdx = rowBase + l;
  if (arowIdx >= nNodes) arowIdx = nNodes - 1;
  const float* __restrict__ arow = A + (size_t)arowIdx * D;

  const int col0 = wave * 32;
  const float* __restrict__ bp0 = W + (col0 + l);        // tile 0, column n=l
  const float* __restrict__ bp1 = W + (col0 + 16 + l);   // tile 1

  v8f acc0 = {};
  v8f acc1 = {};

  for (int k = 0; k < D; k += 4) {
    const int kk = k + 2 * half;
    // A fragment: float2 (K=kk, kk+1) for row M=l — 8B-aligned
    v2f a = *(const v2f*)(arow + kk);
    // B fragments: rows kk, kk+1 of W at this wave's columns (stride D)
    v2f b0; b0.x = bp0[(size_t)kk * D]; b0.y = bp0[(size_t)(kk + 1) * D];
    v2f b1; b1.x = bp1[(size_t)kk * D]; b1.y = bp1[(size_t)(kk + 1) * D];

    acc0 = __builtin_amdgcn_wmma_f32_16x16x4_f32(
        false, a, false, b0, (short)0, acc0, false, false);
    acc1 = __builtin_amdgcn_wmma_f32_16x16x4_f32(
        false, a, false, b1, (short)0, acc1, false, false);
  }

  // C/D layout: VGPR j -> lanes 0-15: (M=j, N=lane); lanes 16-31: (M=8+j, N=lane-16)
#pragma unroll
  for (int j = 0; j < 8; ++j) {
    const int m = rowBase + j + 8 * half;
    if (m < nNodes) {
      float* __restrict__ orow = X + (size_t)m * D;
      orow[col0 + l]      = acc0[j];
      orow[col0 + 16 + l] = acc1[j];
    }
  }
}

// -------------------------------------------------------------------------
// Kernel 2: zero the output accumulator (float4 stores)
// -------------------------------------------------------------------------
__global__ void zero_f4(float4* __restrict__ out, long n4) {
  long i = (long)blockIdx.x * blockDim.x + threadIdx.x;
  if (i < n4) out[i] = make_float4(0.f, 0.f, 0.f, 0.f);
}

// -------------------------------------------------------------------------
// Kernel 3: SpMM scatter-add.  One wave handles 32 edges: edge meta loaded
// coalesced (one edge per lane), broadcast via __shfl; per edge the wave
// moves 256 floats (2x float4 per lane) and issues no-return f32 atomics
// (hit mostly in the 192MB L2 since x and out are ~102MB each).
// -------------------------------------------------------------------------
__global__ __launch_bounds__(256) void spmm_scatter(
    const float* __restrict__ X,
    const int*   __restrict__ esrc,
    const int*   __restrict__ edst,
    const float* __restrict__ eval,
    float* __restrict__ out,
    int nE) {
  const int lane = threadIdx.x & 31;
  const long wave = ((long)blockIdx.x * blockDim.x + threadIdx.x) >> 5;
  const long e0 = wave * 32;
  if (e0 >= nE) return;

  long eL = e0 + lane;
  int s = 0, d = 0; float v = 0.f;
  if (eL < nE) { s = esrc[eL]; d = edst[eL]; v = eval[eL]; }

  long rem = (long)nE - e0;
  const int nIter = rem < 32 ? (int)rem : 32;

  for (int i = 0; i < nIter; ++i) {
    const int   si = __shfl(s, i);
    const int   di = __shfl(d, i);
    const float vi = __shfl(v, i);

    const float4* __restrict__ xs = (const float4*)(X + (size_t)si * D);
    const float4 a = xs[lane];        // cols [lane*4, lane*4+4)
    const float4 b = xs[lane + 32];   // cols [128+lane*4, ...)

    float* __restrict__ od = out + (size_t)di * D;
    atomicAdd(od + lane * 4 + 0, vi * a.x);
    atomicAdd(od + lane * 4 + 1, vi * a.y);
    atomicAdd(od + lane * 4 + 2, vi * a.z);
    atomicAdd(od + lane * 4 + 3, vi * a.w);
    atomicAdd(od + 128 + lane * 4 + 0, vi * b.x);
    atomicAdd(od + 128 + lane * 4 + 1, vi * b.y);
    atomicAdd(od + 128 + lane * 4 + 2, vi * b.z);
    atomicAdd(od + 128 + lane * 4 + 3, vi * b.w);
  }
}

// -------------------------------------------------------------------------
// Kernel 4: in-place ReLU
// -------------------------------------------------------------------------
__global__ void relu_inplace(float* __restrict__ out, long n) {
  long i = (long)blockIdx.x * blockDim.x + threadIdx.x;
  if (i < n) out[i] = fmaxf(out[i], 0.f);
}

// -------------------------------------------------------------------------
extern "C" void kernel_launch(void* const* d_in, const int* in_sizes, int n_in,
                              void* d_out, int out_size, void* d_ws, size_t ws_size,
                              hipStream_t stream) {
  const float* inputs = (const float*)d_in[0];   // [N, 256] f32
  const float* weight = (const float*)d_in[1];   // [256, 256] f32
  const int*   esrc   = (const int*)d_in[2];     // [E] i32
  const int*   edst   = (const int*)d_in[3];     // [E] i32
  const float* evals  = (const float*)d_in[4];   // [E] f32
  float* out = (float*)d_out;

  const int nNodes = in_sizes[0] / D;            // 100000
  const int nE     = in_sizes[2];                // 3200000

  float* X = (float*)d_ws;                       // [nNodes, 256] scratch (102.4 MB)

  // 1) dense GEMM: x = inputs @ W  (16-row stripes; 100000/16 = 6250 exact)
  const int gemmBlocks = (nNodes + 15) / 16;
  gemm_f32_wmma<<<gemmBlocks, 256, 0, stream>>>(inputs, weight, X, nNodes);

  // 2) zero output accumulator
  const long n = (long)out_size;
  zero_f4<<<(unsigned)((n / 4 + 255) / 256), 256, 0, stream>>>((float4*)out, n / 4);

  // 3) edge-parallel scatter-add (8 waves/block, 32 edges per wave)
  const long nWaves  = ((long)nE + 31) / 32;
  const long nBlocks = (nWaves + 7) / 8;
  spmm_scatter<<<(unsigned)nBlocks, 256, 0, stream>>>(X, esrc, edst, evals, out, nE);

  // 4) ReLU
  relu_inplace<<<(unsigned)((n + 255) / 256), 256, 0, stream>>>(out, n);
}